// SelfAttention_42090679500836
// MI455X (gfx1250) — compile-verified
//
#include <hip/hip_runtime.h>
#include <math.h>

typedef __attribute__((ext_vector_type(2))) float v2f;
typedef __attribute__((ext_vector_type(4))) float v4f;
typedef __attribute__((ext_vector_type(8))) float v8f;

// D = A(16x4) * B(4x16) + C, fp32, wave32.
#define WMMA_F32(a, b, c) \
  __builtin_amdgcn_wmma_f32_16x16x4_f32(false, (a), false, (b), (short)0, (c), false, false)

// ---------------------------------------------------------------------------
// GEMM + bias: C[M,N] = A[M,512] * W[512,N] + bias[N], N compile-time.
// Block = 128 threads (4 waves) -> 64x64 C tile; wave tile = 16(M) x 64(N).
// LDS double-buffered K-slabs of 32; global loads overlapped with WMMA.
//   As: [64][36]  padded  -> conflict-free A-fragment v2f reads
//   Wq: [8][64][4] quad-k -> B fragment = one ds_load_b64, banks disjoint
//       Wq[k>>2][n] = { W[k0][n], W[k0+1][n], W[k0+2][n], W[k0+3][n] }
// ---------------------------------------------------------------------------
template <int N>
__global__ __launch_bounds__(128) void gemm_bias_f32(
    const float* __restrict__ A, const float* __restrict__ W,
    const float* __restrict__ bias, float* __restrict__ C)
{
  constexpr int K  = 512;
  constexpr int KS = 32;            // K-slab
  constexpr int NSLAB = K / KS;     // 16

  __shared__ float As[2][64][36];
  __shared__ float Wq[2][KS / 4][64][4];

  const int t      = threadIdx.x;
  const int lane   = t & 31;
  const int wave   = t >> 5;
  const int lane16 = lane & 15;
  const int hi     = lane >> 4;     // 0: K rows {0,1}; 1: K rows {2,3}
  const int mBase  = blockIdx.y * 64;
  const int nBase  = blockIdx.x * 64;

  // staging thread mapping
  const int aRow = t >> 3;          // 0..15 (+16 per pass)
  const int aCol = (t & 7) * 4;     // 0..28
  const int wRow = t >> 4;          // 0..7  (+8 per pass)
  const int wCol = (t & 15) * 4;    // 0..60

  const float* Ag = A + (size_t)mBase * K;
  const float* Wg = W + nBase;

  v4f areg[4], wreg[4];
  auto loadSlab = [&](int s) {
    const int k0 = s * KS;
    #pragma unroll
    for (int p = 0; p < 4; ++p) {
      areg[p] = *(const v4f*)(Ag + (size_t)(p * 16 + aRow) * K + k0 + aCol);
      wreg[p] = *(const v4f*)(Wg + (size_t)(k0 + p * 8 + wRow) * N + wCol);
    }
  };
  auto storeSlab = [&](int buf) {
    #pragma unroll
    for (int p = 0; p < 4; ++p) {
      *(v4f*)&As[buf][p * 16 + aRow][aCol] = areg[p];
      const int kr = p * 8 + wRow;
      #pragma unroll
      for (int i = 0; i < 4; ++i)
        Wq[buf][kr >> 2][wCol + i][kr & 3] = wreg[p][i];
    }
  };

  loadSlab(0);
  storeSlab(0);

  v8f acc[4] = {};
  for (int s = 0; s < NSLAB; ++s) {
    if (s + 1 < NSLAB) loadSlab(s + 1);   // issue next slab's global loads
    __syncthreads();                      // current slab visible in LDS
    const int buf = s & 1;
    #pragma unroll
    for (int ks = 0; ks < KS; ks += 4) {
      // A fragment: rows m = wave*16+lane16, k = ks + 2*hi (+1)
      v2f a = *(const v2f*)&As[buf][wave * 16 + lane16][ks + 2 * hi];
      #pragma unroll
      for (int j = 0; j < 4; ++j) {
        v2f bf = *(const v2f*)&Wq[buf][ks >> 2][j * 16 + lane16][2 * hi];
        acc[j] = WMMA_F32(a, bf, acc[j]);
      }
    }
    if (s + 1 < NSLAB) storeSlab((s + 1) & 1);  // fill other buffer
  }

  #pragma unroll
  for (int j = 0; j < 4; ++j) {
    const int n  = nBase + j * 16 + lane16;
    const float bv = bias[n];
    #pragma unroll
    for (int r = 0; r < 8; ++r) {
      const int m = mBase + wave * 16 + r + hi * 8;
      C[(size_t)m * N + n] = acc[j][r] + bv;
    }
  }
}

// ---------------------------------------------------------------------------
// Causal flash attention, fp32, one block per (b, h, 64-row q tile).
// qkv layout: [b*S + s][1536]: cols [0,512)=Q, [512,1024)=K, [1024,1536)=V,
// head h at cols h*64..h*64+63 inside each third.
// Next K/V tile prefetched into registers while current tile computes.
// ---------------------------------------------------------------------------
__global__ __launch_bounds__(128) void attention_f32(
    const float* __restrict__ qkv, float* __restrict__ out)
{
  constexpr int S      = 2048;
  constexpr int DMODEL = 512;
  constexpr int QKVLD  = 1536;
  constexpr int HD     = 64;
  constexpr float SCALE = 0.125f;  // 1/sqrt(64)

  __shared__ float Kt[64][68];       // padded: conflict-free, 16B-aligned rows
  __shared__ float Vt[64][68];
  __shared__ float Pt[4][16][68];    // per-wave P tile (C-layout -> A-layout)

  const int lane   = threadIdx.x & 31;
  const int wave   = threadIdx.x >> 5;
  const int lane16 = lane & 15;
  const int hi     = lane >> 4;
  const int qTile  = blockIdx.x;     // 0..31
  const int h      = blockIdx.y;     // 0..7
  const int b      = blockIdx.z;     // 0..3
  const int qBase  = qTile * 64 + wave * 16;

  // Preload this wave's Q rows as 16 A-fragments (full K=64 in steps of 4).
  const float* Qp = qkv + (size_t)(b * S + qBase + lane16) * QKVLD + h * HD;
  v2f qreg[16];
  #pragma unroll
  for (int i = 0; i < 16; ++i)
    qreg[i] = *(const v2f*)(Qp + 4 * i + 2 * hi);

  v8f oacc[4] = {};
  float mrow[8], lrow[8];
  #pragma unroll
  for (int r = 0; r < 8; ++r) { mrow[r] = -1e30f; lrow[r] = 0.0f; }

  const int ldRow = threadIdx.x >> 4;        // 0..7
  const int ldCol = (threadIdx.x & 15) * 4;  // 0,4,...,60

  // K/V tile register prefetch (64x64 each: 8 v4f per thread per tile)
  v4f kreg[8], vreg[8];
  auto loadTile = [&](int kt) {
    const int kBase = kt * 64;
    #pragma unroll
    for (int p = 0; p < 8; ++p) {
      const int row = p * 8 + ldRow;
      const float* Ksrc =
          qkv + (size_t)(b * S + kBase + row) * QKVLD + DMODEL + h * HD + ldCol;
      kreg[p] = *(const v4f*)Ksrc;
      vreg[p] = *(const v4f*)(Ksrc + DMODEL);
    }
  };

  loadTile(0);

  for (int kt = 0; kt <= qTile; ++kt) {
    const int kBase = kt * 64;
    __syncthreads();                         // all waves done reading prev tile
    #pragma unroll
    for (int p = 0; p < 8; ++p) {
      const int row = p * 8 + ldRow;
      *(v4f*)&Kt[row][ldCol] = kreg[p];
      *(v4f*)&Vt[row][ldCol] = vreg[p];
    }
    if (kt < qTile) loadTile(kt + 1);        // overlap next tile with compute
    __syncthreads();                         // tile visible

    // ---- scores: sc[j] (16 q rows x 16 keys) = Q * K^T ----
    v8f sc[4] = {};
    #pragma unroll
    for (int i = 0; i < 16; ++i) {
      const int d0 = 4 * i + 2 * hi;
      v2f a = qreg[i];
      #pragma unroll
      for (int j = 0; j < 4; ++j) {
        v2f bf = *(const v2f*)&Kt[j * 16 + lane16][d0];  // lane = key
        sc[j] = WMMA_F32(a, bf, sc[j]);
      }
    }

    // ---- online softmax (each row lives in one 16-lane half) ----
    const bool diag = (kt == qTile);
    #pragma unroll
    for (int r = 0; r < 8; ++r) {
      const int q = qBase + r + hi * 8;
      float rm = -1e30f;
      #pragma unroll
      for (int j = 0; j < 4; ++j) {
        float sv = sc[j][r] * SCALE;
        if (diag && (kBase + j * 16 + lane16) > q) sv = -1e30f;
        sc[j][r] = sv;
        rm = fmaxf(rm, sv);
      }
      #pragma unroll
      for (int off = 1; off < 16; off <<= 1)
        rm = fmaxf(rm, __shfl_xor(rm, off));
      const float mn    = fmaxf(mrow[r], rm);
      const float alpha = __expf(mrow[r] - mn);
      float rs = 0.0f;
      #pragma unroll
      for (int j = 0; j < 4; ++j) {
        float pv = __expf(sc[j][r] - mn);
        Pt[wave][r + hi * 8][j * 16 + lane16] = pv;
        rs += pv;
      }
      #pragma unroll
      for (int off = 1; off < 16; off <<= 1)
        rs += __shfl_xor(rs, off);
      lrow[r] = lrow[r] * alpha + rs;
      mrow[r] = mn;
      #pragma unroll
      for (int j = 0; j < 4; ++j)
        oacc[j][r] *= alpha;
    }

    // ---- O += P(16x64) * V(64x64); P re-read from LDS in A-layout ----
    #pragma unroll
    for (int i = 0; i < 16; ++i) {
      const int kk = 4 * i + 2 * hi;
      v2f a = *(const v2f*)&Pt[wave][lane16][kk];
      #pragma unroll
      for (int j = 0; j < 4; ++j) {
        v2f bf;
        bf.x = Vt[kk][j * 16 + lane16];
        bf.y = Vt[kk + 1][j * 16 + lane16];
        oacc[j] = WMMA_F32(a, bf, oacc[j]);
      }
    }
  }

  // ---- normalize and store in (b, s, h*64+dd) layout ----
  float* Op = out + (size_t)(b * S) * DMODEL + h * HD;
  #pragma unroll
  for (int r = 0; r < 8; ++r) {
    const float inv = 1.0f / lrow[r];
    const int q = qBase + r + hi * 8;
    #pragma unroll
    for (int j = 0; j < 4; ++j)
      Op[(size_t)q * DMODEL + j * 16 + lane16] = oacc[j][r] * inv;
  }
}

// ---------------------------------------------------------------------------
extern "C" void kernel_launch(void* const* d_in, const int* in_sizes, int n_in,
                              void* d_out, int out_size, void* d_ws, size_t ws_size,
                              hipStream_t stream) {
  (void)in_sizes; (void)n_in; (void)out_size; (void)ws_size;

  const float* x     = (const float*)d_in[0];  // (4, 2048, 512)
  const float* w_qkv = (const float*)d_in[1];  // (512, 1536)
  const float* b_qkv = (const float*)d_in[2];  // (1536,)
  const float* w_out = (const float*)d_in[3];  // (512, 512)
  const float* b_out = (const float*)d_in[4];  // (512,)
  float* outp = (float*)d_out;                 // (4, 2048, 512)

  constexpr int M = 4 * 2048;   // 8192 rows
  constexpr int D = 512;
  constexpr int QKVN = 3 * D;   // 1536

  float* qkv  = (float*)d_ws;                 // 8192*1536 f32 = 48 MB
  float* attn = qkv + (size_t)M * QKVN;       // 8192*512  f32 = 16 MB

  // 1) QKV projection
  gemm_bias_f32<QKVN><<<dim3(QKVN / 64, M / 64), 128, 0, stream>>>(
      x, w_qkv, b_qkv, qkv);

  // 2) causal flash attention per (b, h, q-tile)
  attention_f32<<<dim3(2048 / 64, 8, 4), 128, 0, stream>>>(qkv, attn);

  // 3) output projection
  gemm_bias_f32<D><<<dim3(D / 64, M / 64), 128, 0, stream>>>(
      attn, w_out, b_out, outp);
}